// KuramotoLayer_60224031425028
// MI455X (gfx1250) — compile-verified
//
#include <hip/hip_runtime.h>
#include <cmath>

// Kuramoto: theta_{t+1} = theta_t + DT*(omega + cos*(K@sin) - sin*(K@cos))
// K symmetric => K@x == x^T K, so both matvecs are a single skinny GEMM
// [s;c](2x4096) x K(4096x4096) via V_WMMA_F32_16X16X4_F32, reading K
// exactly once per step (64MB, resident in the 192MB L2).
//
// B-operand is fetched as b128: lane n holds K[row][n0+4n .. n0+4n+3] and
// 4 WMMAs per row-pair compute the 4 interleaved column subsets, so vmem
// instruction cost is 0.5 loads per WMMA at full 256B/half-wave coalescing.

typedef float v2f __attribute__((ext_vector_type(2)));
typedef float v4f __attribute__((ext_vector_type(4)));
typedef float v8f __attribute__((ext_vector_type(8)));

constexpr int   KN      = 4096;
constexpr int   NSTEPS  = 50;
constexpr float DT      = 0.01f;

constexpr int   TPB     = 1024;         // threads per block (32 wave32)
constexpr int   WPB     = TPB / 32;     // waves per block
constexpr int   OPB     = 64;           // oscillators per block
constexpr int   KSLAB   = KN / WPB;     // 128 k's per wave
constexpr int   NGRP    = KSLAB / 4;    // 32 k-groups (4 k's each) per wave

// ---------------------------------------------------------------- init ----
__global__ void init_theta(const float* __restrict__ phases,
                           float* __restrict__ theta) {
    int i = blockIdx.x * blockDim.x + threadIdx.x;
    if (i < KN) theta[i] = phases[i];
}

// --------------------------------------------------------------- sincos ---
__global__ void phase_sincos(const float* __restrict__ theta,
                             float* __restrict__ s,
                             float* __restrict__ c) {
    int i = blockIdx.x * blockDim.x + threadIdx.x;
    if (i < KN) {
        float sv, cv;
        sincosf(theta[i], &sv, &cv);
        s[i] = sv;
        c[i] = cv;
    }
}

// ---------------------------------------------------- fused matvec step ---
// WMMA operand striping (ISA 7.12.2, wave32):
//   A (16x4 f32, 2 VGPR): lane L: m = L&15, h = L>>4; a[j] = A[m][2h+j]
//   B (4x16 f32, 2 VGPR): lane L: n = L&15, h = L>>4; b[j] = B[2h+j][n]
//   D (16x16 f32, 8 VGPR): VGPR v, lanes0-15: D[v][lane]
// A row0 = s-slice, row1 = c-slice, rows 2..15 = 0.
// WMMA q over component q of the b128 => D_q[0][n] = (K@s)[n0+4n+q],
// D_q[1][n] = (K@c)[n0+4n+q] in lanes 0-15.
__global__ __launch_bounds__(TPB)
void kuramoto_step(const float* __restrict__ K,
                   const float* __restrict__ s,
                   const float* __restrict__ c,
                   const float* __restrict__ omegas,
                   float* __restrict__ theta) {
    __shared__ float lds_s[KN];
    __shared__ float lds_c[KN];
    __shared__ float red_ys[WPB][OPB];
    __shared__ float red_yc[WPB][OPB];

    const int tid = threadIdx.x;

    // Stage the full sin/cos vectors (32 KB) — shared by all 32 waves.
    for (int i = tid; i < KN; i += TPB) {
        lds_s[i] = s[i];
        lds_c[i] = c[i];
    }
    __syncthreads();

    const int n0   = blockIdx.x * OPB;  // 64-oscillator output tile
    const int wave = tid >> 5;
    const int lane = tid & 31;
    const int m    = lane & 15;
    const int h    = lane >> 4;

    const int k0 = wave * KSLAB;        // 128 k's per wave

    v8f acc0 = {}, acc1 = {}, acc2 = {}, acc3 = {};

    // Per-lane B pointer: row (k + 2h + j), cols [n0+4m, n0+4m+4); j=1 is +KN.
    const float* bptr = K + (size_t)(k0 + 2 * h) * KN + (n0 + 4 * m);

    // Per-lane A source: unconditional LDS float2 load, then lane-mask.
    const float* aptr = ((m == 0) ? lds_s : lds_c) + 2 * h;
    const bool   a_on = (m < 2);

    // ---- software pipeline: prologue group load ----
    v2f a_c;
    v4f b0_c, b1_c;
    {
        v2f av = *(const v2f*)(aptr + k0);        // ds_load_b64
        a_c.x = a_on ? av.x : 0.0f;
        a_c.y = a_on ? av.y : 0.0f;
        b0_c  = *(const v4f*)(bptr);              // global_load_b128 row k+2h
        b1_c  = *(const v4f*)(bptr + KN);         // global_load_b128 row k+2h+1
        bptr += (size_t)4 * KN;
    }

    #pragma unroll 4
    for (int g = 1; g < NGRP; ++g) {
        const int k = k0 + 4 * g;

        // issue next group's loads before consuming current operands
        v2f av = *(const v2f*)(aptr + k);
        v2f a_n;
        a_n.x = a_on ? av.x : 0.0f;
        a_n.y = a_on ? av.y : 0.0f;
        v4f b0_n = *(const v4f*)(bptr);
        v4f b1_n = *(const v4f*)(bptr + KN);
        bptr += (size_t)4 * KN;

        v2f b;
        b.x = b0_c.x; b.y = b1_c.x;
        acc0 = __builtin_amdgcn_wmma_f32_16x16x4_f32(false, a_c, false, b,
                                                     (short)0, acc0, false, false);
        b.x = b0_c.y; b.y = b1_c.y;
        acc1 = __builtin_amdgcn_wmma_f32_16x16x4_f32(false, a_c, false, b,
                                                     (short)0, acc1, false, false);
        b.x = b0_c.z; b.y = b1_c.z;
        acc2 = __builtin_amdgcn_wmma_f32_16x16x4_f32(false, a_c, false, b,
                                                     (short)0, acc2, false, false);
        b.x = b0_c.w; b.y = b1_c.w;
        acc3 = __builtin_amdgcn_wmma_f32_16x16x4_f32(false, a_c, false, b,
                                                     (short)0, acc3, false, false);

        a_c  = a_n;
        b0_c = b0_n;
        b1_c = b1_n;
    }

    // ---- epilogue group ----
    {
        v2f b;
        b.x = b0_c.x; b.y = b1_c.x;
        acc0 = __builtin_amdgcn_wmma_f32_16x16x4_f32(false, a_c, false, b,
                                                     (short)0, acc0, false, false);
        b.x = b0_c.y; b.y = b1_c.y;
        acc1 = __builtin_amdgcn_wmma_f32_16x16x4_f32(false, a_c, false, b,
                                                     (short)0, acc1, false, false);
        b.x = b0_c.z; b.y = b1_c.z;
        acc2 = __builtin_amdgcn_wmma_f32_16x16x4_f32(false, a_c, false, b,
                                                     (short)0, acc2, false, false);
        b.x = b0_c.w; b.y = b1_c.w;
        acc3 = __builtin_amdgcn_wmma_f32_16x16x4_f32(false, a_c, false, b,
                                                     (short)0, acc3, false, false);
    }

    // Cross-wave reduction: lane n owns oscillators n0+4n+q (q=0..3).
    if (lane < 16) {
        red_ys[wave][4 * lane + 0] = acc0[0];
        red_ys[wave][4 * lane + 1] = acc1[0];
        red_ys[wave][4 * lane + 2] = acc2[0];
        red_ys[wave][4 * lane + 3] = acc3[0];
        red_yc[wave][4 * lane + 0] = acc0[1];
        red_yc[wave][4 * lane + 1] = acc1[1];
        red_yc[wave][4 * lane + 2] = acc2[1];
        red_yc[wave][4 * lane + 3] = acc3[1];
    }
    __syncthreads();

    if (tid < OPB) {
        float ys = 0.0f, yc = 0.0f;
        #pragma unroll
        for (int w = 0; w < WPB; ++w) {
            ys += red_ys[w][tid];
            yc += red_yc[w][tid];
        }
        const int   i  = n0 + tid;
        const float si = lds_s[i];
        const float ci = lds_c[i];
        theta[i] = theta[i] + DT * (omegas[i] + ci * ys - si * yc);
    }
}

// ---------------------------------------------------------------- launch --
extern "C" void kernel_launch(void* const* d_in, const int* in_sizes, int n_in,
                              void* d_out, int out_size, void* d_ws, size_t ws_size,
                              hipStream_t stream) {
    (void)in_sizes; (void)n_in; (void)out_size; (void)ws_size;

    const float* phases = (const float*)d_in[0];
    const float* K      = (const float*)d_in[1];
    const float* omegas = (const float*)d_in[2];

    float* theta = (float*)d_out;          // theta state lives in d_out
    float* s     = (float*)d_ws;           // 4096 floats
    float* c     = s + KN;                 // 4096 floats

    init_theta<<<KN / 256, 256, 0, stream>>>(phases, theta);

    for (int step = 0; step < NSTEPS; ++step) {
        phase_sincos<<<KN / 256, 256, 0, stream>>>(theta, s, c);
        kuramoto_step<<<KN / OPB, TPB, 0, stream>>>(K, s, c, omegas, theta);
    }
}